// MultiHeadAttention_60181081752214
// MI455X (gfx1250) — compile-verified
//
#include <hip/hip_runtime.h>
#include <hip/hip_bf16.h>

// Problem constants (from reference): B=2, T=2048, E=1024, H=16, D=E/H=64
constexpr int Bc = 2;
constexpr int Tc = 2048;
constexpr int Ec = 1024;
constexpr int Hc = 16;
constexpr int Dc = 64;
constexpr int LDSW = 68;   // padded LDS row stride (floats); 272B = 17*16 -> B128-aligned
constexpr int TILEF = 16 * LDSW;  // one 16x64 staged tile (padded), in floats

typedef float v2f __attribute__((ext_vector_type(2)));
typedef float v8f __attribute__((ext_vector_type(8)));

// b128 async copy operand types (match builtin signature: v4i pointers)
typedef int v4i __attribute__((vector_size(16)));
typedef v4i __attribute__((address_space(1))) gv4i_t;  // global
typedef v4i __attribute__((address_space(3))) lv4i_t;  // LDS

#ifndef __has_builtin
#define __has_builtin(x) 0
#endif

#if __has_builtin(__builtin_amdgcn_global_load_async_to_lds_b128)
#define HAVE_ASYNC 1
#else
#define HAVE_ASYNC 0
#endif

#if HAVE_ASYNC
#if __has_builtin(__builtin_amdgcn_s_wait_asynccnt)
#define ASYNC_WAIT(n) __builtin_amdgcn_s_wait_asynccnt(n)
#else
#define ASYNC_WAIT(n) asm volatile("s_wait_asynccnt " #n ::: "memory")
#endif
#else
#define ASYNC_WAIT(n) ((void)0)
#endif

__device__ inline v8f wmma4(v2f a, v2f b, v8f c) {
  // D = A(16x4 f32) x B(4x16 f32) + C(16x16 f32)
  return __builtin_amdgcn_wmma_f32_16x16x4_f32(false, a, false, b, (short)0, c,
                                               false, false);
}

__device__ inline float rowmax16(float v) {
  v = fmaxf(v, __shfl_xor(v, 1));
  v = fmaxf(v, __shfl_xor(v, 2));
  v = fmaxf(v, __shfl_xor(v, 4));
  v = fmaxf(v, __shfl_xor(v, 8));
  return v;
}
__device__ inline float rowsum16(float v) {
  v += __shfl_xor(v, 1);
  v += __shfl_xor(v, 2);
  v += __shfl_xor(v, 4);
  v += __shfl_xor(v, 8);
  return v;
}

// Load 16 k-step fragments (v2f each) from an LDS row: elements [row][k],[row][k+1]
// with k = step*4 + half*2.  Batched so the ds_loads group and WMMAs chain.
__device__ inline void load_frags16(v2f* f, const float* base, int row, int half) {
#pragma unroll
  for (int step = 0; step < 16; ++step) {
    int k = step * 4 + half * 2;
    f[step].x = base[row * LDSW + k];
    f[step].y = base[row * LDSW + k + 1];
  }
}

#if HAVE_ASYNC
__device__ inline void async_b128(const float* g, float* l) {
  __builtin_amdgcn_global_load_async_to_lds_b128((gv4i_t*)g, (lv4i_t*)l, 0, 0);
}
#endif

// Wave-cooperative copy of a 16x64 f32 tile (global row stride = gstride floats)
// into LDS with row stride LDSW. Async B128 path (ASYNCcnt) when available.
__device__ inline void issue_tile_copy_wave(float* lds, const float* g,
                                            int gstride, int lane) {
#if HAVE_ASYNC
  // ensure in-flight DS ops on this region from >=2 iterations ago are done
  asm volatile("s_wait_dscnt 0x0" ::: "memory");
#pragma unroll
  for (int it = 0; it < 8; ++it) {
    int c = lane + it * 32;  // 256 chunks of 4 floats
    int row = c >> 4, col = (c & 15) << 2;
    async_b128(g + (size_t)row * gstride + col, lds + row * LDSW + col);
  }
#else
  for (int i = lane; i < 16 * 64; i += 32)
    lds[(i >> 6) * LDSW + (i & 63)] = g[(size_t)(i >> 6) * gstride + (i & 63)];
#endif
}

// Block-cooperative (256 threads) copy of ROWS x 64 f32 tile into LDS.
template <int ROWS>
__device__ inline void issue_tile_copy_block(float* lds, const float* g,
                                             int gstride, int tid) {
#if HAVE_ASYNC
#pragma unroll
  for (int it = 0; it < (ROWS * 16) / 256; ++it) {
    int c = tid + it * 256;
    int row = c >> 4, col = (c & 15) << 2;
    async_b128(g + (size_t)row * gstride + col, lds + row * LDSW + col);
  }
#else
  for (int i = tid; i < ROWS * 64; i += 256)
    lds[(i >> 6) * LDSW + (i & 63)] = g[(size_t)(i >> 6) * gstride + (i & 63)];
#endif
}

// ---------------------------------------------------------------------------
// Kernel 1: QKV projection.  Q/K/V[b,h,t,d] = sum_e x[b,t,e] * W{q,k,v}[h,e,d]
// One block per (b,h,t-tile of 16). 256 threads = 8 waves.
// Wave w owns output tiles {w} and (w<4) {w+8}; tile = mat*4 + dtile.
// ---------------------------------------------------------------------------
__global__ __launch_bounds__(256) void qkv_kernel(
    const float* __restrict__ x, const float* __restrict__ Wq,
    const float* __restrict__ Wk, const float* __restrict__ Wv,
    float* __restrict__ Qo, float* __restrict__ Ko, float* __restrict__ Vo) {
  __shared__ float xs[16 * LDSW];
  __shared__ float wt[3][64][LDSW];  // W^T chunks: wt[mat][d][k]

  const int tid = threadIdx.x;
  const int lane = tid & 31;
  const int wv = __builtin_amdgcn_readfirstlane(tid >> 5);  // scalar wave id
  const int half = lane >> 4, l16 = lane & 15;
  const int tt = blockIdx.x & (Tc / 16 - 1);
  const int bh = blockIdx.x >> 7;  // T/16 = 128 tiles
  const int b = bh >> 4, h = bh & (Hc - 1);
  const int t0 = tt * 16;

  const float* Wm[3] = {Wq, Wk, Wv};

  const int mat0 = wv >> 2, dt0 = wv & 3;  // scalar
  const bool has1 = (wv < 4);              // scalar -> s_cbranch, EXEC untouched
  float* O0 = (mat0 == 0) ? Qo : ((mat0 == 1) ? Ko : Vo);
  float* O1 = Vo;  // tiles 8..11 are V

  v8f acc0 = {0, 0, 0, 0, 0, 0, 0, 0};
  v8f acc1 = {0, 0, 0, 0, 0, 0, 0, 0};

  for (int kc = 0; kc < Ec / 64; ++kc) {
    issue_tile_copy_block<16>(xs, x + ((size_t)b * Tc + t0) * Ec + kc * 64, Ec, tid);
    for (int m = 0; m < 3; ++m) {  // transposed stage (gather) stays on VGPR path
      const float* Wp = Wm[m] + ((size_t)h * Ec + kc * 64) * Dc;
      for (int i = tid; i < 64 * 64; i += 256) {
        int k = i >> 6, d = i & 63;
        wt[m][d][k] = Wp[(size_t)k * Dc + d];
      }
    }
    ASYNC_WAIT(0);
    __syncthreads();

    v2f af[16];
    load_frags16(af, xs, l16, half);
    {
      v2f bf[16];
      load_frags16(bf, &wt[mat0][0][0], dt0 * 16 + l16, half);
#pragma unroll
      for (int step = 0; step < 16; ++step) acc0 = wmma4(af[step], bf[step], acc0);
    }
    if (has1) {  // hoisted scalar branch: one test per chunk
      v2f bf[16];
      load_frags16(bf, &wt[2][0][0], dt0 * 16 + l16, half);
#pragma unroll
      for (int step = 0; step < 16; ++step) acc1 = wmma4(af[step], bf[step], acc1);
    }
    __syncthreads();
  }
  for (int r = 0; r < 8; ++r) {
    int m = r + half * 8;
    size_t o = ((size_t)bh * Tc + t0 + m) * Dc;
    O0[o + dt0 * 16 + l16] = acc0[r];
    if (has1) O1[o + dt0 * 16 + l16] = acc1[r];
  }
}

// ---------------------------------------------------------------------------
// Kernel 2: causal ALiBi attention for one (b,h,16-query tile).
// Pass 1: per-wave online softmax stats over wave-strided 16x16 S-tiles with
//         double-buffered async K staging. Combine stats across waves in LDS.
// Pass 2: recompute S, write normalized probabilities (zeros above the causal
//         bound; d_out is poisoned), accumulate ctx = P @ V via WMMA,
//         deterministic fixed-order cross-wave ctx reduction.
// Q-side A-fragments are loaded into registers once and reused by every WMMA.
// ---------------------------------------------------------------------------
__global__ __launch_bounds__(256) void attn_kernel(
    const float* __restrict__ Qi, const float* __restrict__ Ki,
    const float* __restrict__ Vi, float* __restrict__ attn,
    float* __restrict__ ctxo) {
  extern __shared__ float smem[];
  float* qs = smem;                       // TILEF
  float* ks = qs + TILEF;                 // 8 waves * 2 buffers * TILEF
  float* mw = ks + 8 * 2 * TILEF;         // 8*16
  float* lw = mw + 8 * 16;                // 8*16
  float* rowm = lw + 8 * 16;              // 16
  float* rowli = rowm + 16;               // 16
  float* ctxp = rowli + 16;               // 8*16*64

  const int tid = threadIdx.x;
  const int lane = tid & 31;
  const int wv = __builtin_amdgcn_readfirstlane(tid >> 5);  // scalar wave id
  const int half = lane >> 4, l16 = lane & 15;
  const int tt = blockIdx.x & (Tc / 16 - 1);
  const int bh = blockIdx.x >> 7;
  const int h = bh & (Hc - 1);
  const int t0 = tt * 16;
  const float slope = exp2f(-0.5f * (float)(h + 1));  // (2^(-8/H))^(h+1)
  const float scale = 0.125f;                          // D^-0.5
  const float NEGBIG = -1e30f;

  const float* Qp = Qi + (size_t)bh * Tc * Dc;
  const float* Kp = Ki + (size_t)bh * Tc * Dc;
  const float* Vp = Vi + (size_t)bh * Tc * Dc;
  float* Ap = attn + (size_t)bh * Tc * Tc;

  issue_tile_copy_block<16>(qs, Qp + (size_t)t0 * Dc, Dc, tid);
  ASYNC_WAIT(0);
  __syncthreads();

  // Q A-fragments: loop-invariant across all S-tiles and both passes
  v2f aq[16];
  load_frags16(aq, qs, l16, half);

  float* b0 = ks + wv * 2 * TILEF;
  float* b1 = b0 + TILEF;
  const int nt = (tt >= wv) ? ((tt - wv) / 8 + 1) : 0;  // full S-tiles this wave

  float run_m[8], run_l[8];
  for (int r = 0; r < 8; ++r) {
    run_m[r] = NEGBIG;
    run_l[r] = 0.f;
  }

  // ---- pass 1: row stats (double-buffered K staging) ----
  if (nt > 0) issue_tile_copy_wave(b0, Kp + (size_t)(wv * 16) * Dc, Dc, lane);
  for (int idx = 0; idx < nt; ++idx) {
    int st = wv + 8 * idx;
    int s0 = st * 16;
    float* cur = (idx & 1) ? b1 : b0;
    if (idx + 1 < nt) {
      issue_tile_copy_wave((idx & 1) ? b0 : b1,
                           Kp + (size_t)((st + 8) * 16) * Dc, Dc, lane);
      ASYNC_WAIT(8);  // in-order completion: oldest 8 (== cur tile) are done
    } else {
      ASYNC_WAIT(0);
    }
    v2f bk[16];
    load_frags16(bk, cur, l16, half);
    v8f acc = {0, 0, 0, 0, 0, 0, 0, 0};
#pragma unroll
    for (int step = 0; step < 16; ++step) acc = wmma4(aq[step], bk[step], acc);

    for (int r = 0; r < 8; ++r) {
      int i = t0 + r + half * 8, j = s0 + l16;
      float s = (j <= i) ? (acc[r] * scale + slope * (float)(j - i)) : NEGBIG;
      float nm = fmaxf(run_m[r], rowmax16(s));
      float e = (j <= i) ? __expf(s - nm) : 0.f;
      float esum = rowsum16(e);
      run_l[r] = run_l[r] * __expf(run_m[r] - nm) + esum;
      run_m[r] = nm;
    }
  }

  if (l16 == 0) {
    for (int r = 0; r < 8; ++r) {
      mw[wv * 16 + r + half * 8] = run_m[r];
      lw[wv * 16 + r + half * 8] = run_l[r];
    }
  }
  __syncthreads();
  if (tid < 16) {
    float mg = NEGBIG;
    for (int w = 0; w < 8; ++w) mg = fmaxf(mg, mw[w * 16 + tid]);
    float lg = 0.f;
    for (int w = 0; w < 8; ++w) lg += lw[w * 16 + tid] * __expf(mw[w * 16 + tid] - mg);
    rowm[tid] = mg;
    rowli[tid] = 1.f / lg;
  }
  __syncthreads();

  // ---- pass 2: probabilities + ctx (double-buffered K staging) ----
  v8f cacc[4];
  for (int dt = 0; dt < 4; ++dt) {
    v8f z = {0, 0, 0, 0, 0, 0, 0, 0};
    cacc[dt] = z;
  }
  float mgl[8], lil[8];
  for (int r = 0; r < 8; ++r) {
    mgl[r] = rowm[r + half * 8];
    lil[r] = rowli[r + half * 8];
  }

  if (nt > 0) issue_tile_copy_wave(b0, Kp + (size_t)(wv * 16) * Dc, Dc, lane);
  for (int idx = 0; idx < nt; ++idx) {
    int st = wv + 8 * idx;
    int s0 = st * 16;
    float* cur = (idx & 1) ? b1 : b0;
    if (idx + 1 < nt) {
      issue_tile_copy_wave((idx & 1) ? b0 : b1,
                           Kp + (size_t)((st + 8) * 16) * Dc, Dc, lane);
      ASYNC_WAIT(8);
    } else {
      ASYNC_WAIT(0);
    }
    v2f bk[16];
    load_frags16(bk, cur, l16, half);
    v8f acc = {0, 0, 0, 0, 0, 0, 0, 0};
#pragma unroll
    for (int step = 0; step < 16; ++step) acc = wmma4(aq[step], bk[step], acc);

    float p[8];
    for (int r = 0; r < 8; ++r) {
      int i = t0 + r + half * 8, j = s0 + l16;
      p[r] = (j <= i)
                 ? __expf(acc[r] * scale + slope * (float)(j - i) - mgl[r]) * lil[r]
                 : 0.f;
      Ap[(size_t)i * Tc + j] = p[r];
    }
    // re-fragment P as a WMMA A operand via this wave's (consumed) buffer
    for (int r = 0; r < 8; ++r) cur[(r + half * 8) * LDSW + l16] = p[r];
    v2f pf[4];
#pragma unroll
    for (int step = 0; step < 4; ++step) {
      int k = step * 4 + half * 2;
      pf[step].x = cur[l16 * LDSW + k];
      pf[step].y = cur[l16 * LDSW + k + 1];
    }
#pragma unroll
    for (int dt = 0; dt < 4; ++dt) {
      v2f vf[4];
#pragma unroll
      for (int step = 0; step < 4; ++step) {
        int k = step * 4 + half * 2;
        vf[step].x = Vp[((size_t)(s0 + k)) * Dc + dt * 16 + l16];
        vf[step].y = Vp[((size_t)(s0 + k + 1)) * Dc + dt * 16 + l16];
      }
#pragma unroll
      for (int step = 0; step < 4; ++step)
        cacc[dt] = wmma4(pf[step], vf[step], cacc[dt]);
    }
  }
  // strictly above causal bound: attn is exactly zero
  for (int st = wv + 8 * nt; st < Tc / 16; st += 8) {
    int s0 = st * 16;
    for (int r = 0; r < 8; ++r) {
      int i = t0 + r + half * 8, j = s0 + l16;
      Ap[(size_t)i * Tc + j] = 0.f;
    }
  }

  for (int dt = 0; dt < 4; ++dt)
    for (int r = 0; r < 8; ++r)
      ctxp[(wv * 16 + r + half * 8) * 64 + dt * 16 + l16] = cacc[dt][r];
  __syncthreads();
  for (int i = tid; i < 16 * 64; i += 256) {
    int m = i >> 6, d = i & 63;
    float s = 0.f;
    for (int w = 0; w < 8; ++w) s += ctxp[(w * 16 + m) * 64 + d];  // fixed order
    ctxo[((size_t)bh * Tc + t0 + m) * Dc + d] = s;
  }
}

// ---------------------------------------------------------------------------
// Kernel 3: out[b,t,e] = sum_k cat[b,t,k]*Wo[e,k] + bo[e]
// cat[b,t,h*64+d] = ctx[b,h,t,d]; K-chunks of 64 align exactly with heads.
// One block = 16 rows x 128 cols; 8 waves, one 16-col tile each.
// ---------------------------------------------------------------------------
__global__ __launch_bounds__(256) void oproj_kernel(
    const float* __restrict__ ctx, const float* __restrict__ Wo,
    const float* __restrict__ bo, float* __restrict__ out) {
  __shared__ float cs[16 * LDSW];
  __shared__ float wt[128 * LDSW];
  const int tid = threadIdx.x;
  const int lane = tid & 31;
  const int wv = __builtin_amdgcn_readfirstlane(tid >> 5);
  const int half = lane >> 4, l16 = lane & 15;
  const int colBlk = blockIdx.x & 7;  // E/128 = 8
  const int rowTile = blockIdx.x >> 3;
  const int gt0 = rowTile * 16;
  const int b = gt0 / Tc, t0 = gt0 % Tc;
  const int e0 = colBlk * 128;

  v8f acc = {0, 0, 0, 0, 0, 0, 0, 0};
  for (int kc = 0; kc < Ec / 64; ++kc) {  // kc == head index
    issue_tile_copy_block<16>(cs, ctx + (((size_t)b * Hc + kc) * Tc + t0) * Dc,
                              Dc, tid);
    issue_tile_copy_block<128>(wt, Wo + (size_t)e0 * Ec + kc * 64, Ec, tid);
    ASYNC_WAIT(0);
    __syncthreads();

    v2f af[16], bf[16];
    load_frags16(af, cs, l16, half);
    load_frags16(bf, wt, wv * 16 + l16, half);
#pragma unroll
    for (int step = 0; step < 16; ++step) acc = wmma4(af[step], bf[step], acc);
    __syncthreads();
  }
  float bias = bo[e0 + wv * 16 + l16];
  for (int r = 0; r < 8; ++r) {
    int m = r + half * 8;
    out[((size_t)b * Tc + t0 + m) * Ec + e0 + wv * 16 + l16] = acc[r] + bias;
  }
}

extern "C" void kernel_launch(void* const* d_in, const int* in_sizes, int n_in,
                              void* d_out, int out_size, void* d_ws,
                              size_t ws_size, hipStream_t stream) {
  (void)in_sizes; (void)n_in; (void)out_size; (void)ws_size;
  const float* x = (const float*)d_in[0];
  const float* Wq = (const float*)d_in[1];
  const float* Wk = (const float*)d_in[2];
  const float* Wv = (const float*)d_in[3];
  const float* Wo = (const float*)d_in[4];
  const float* bo = (const float*)d_in[5];

  float* out = (float*)d_out;                // [B,T,E]
  float* attn = out + (size_t)Bc * Tc * Ec;  // [B,H,T,T]

  const size_t per = (size_t)Bc * Hc * Tc * Dc;  // 4,194,304 floats
  float* Qb = (float*)d_ws;
  float* Kb = Qb + per;
  float* Vb = Kb + per;
  float* Cb = Vb + per;  // ctx

  const int blocks_bht = Bc * Hc * (Tc / 16);  // 4096
  qkv_kernel<<<blocks_bht, 256, 0, stream>>>(x, Wq, Wk, Wv, Qb, Kb, Vb);

  const size_t shbytes =
      (size_t)(TILEF + 8 * 2 * TILEF + 8 * 16 * 2 + 16 * 2 + 8 * 16 * 64) *
      sizeof(float);  // 107,904 B < 320 KB/WGP
  attn_kernel<<<blocks_bht, 256, shbytes, stream>>>(Qb, Kb, Vb, attn, Cb);

  const int blocks_out = (Bc * Tc / 16) * (Ec / 128);  // 2048
  oproj_kernel<<<blocks_out, 256, 0, stream>>>(Cb, Wo, bo, out);
}